// PositiveDefiniteMatrixGenerator_10917806866515
// MI455X (gfx1250) — compile-verified
//
#include <hip/hip_runtime.h>

typedef __attribute__((ext_vector_type(2))) float v2f;
typedef __attribute__((ext_vector_type(8))) float v8f;

#define BATCH   2048
#define IN_K    1024
#define M_DIM   256
#define P_DIM   32896   // 256*257/2
#define LDS_PAD 20      // 16 data + 4 pad: 16B-aligned rows, conflict-free columns

// Invert packed lower-triangular index p -> (i,j), row-major fill order.
// 8p+1 <= 263161 < 2^24, exact in fp32; integer fixup guards rounding.
__device__ __forceinline__ void tri_inverse(int p, int& i, int& j) {
    int ii = (int)((sqrtf(8.0f * (float)p + 1.0f) - 1.0f) * 0.5f);
    while ((ii + 1) * (ii + 2) / 2 <= p) ++ii;
    while (ii * (ii + 1) / 2 > p) --ii;
    i = ii;
    j = p - ii * (ii + 1) / 2;
}

__device__ __forceinline__ v8f wmma4(v2f a, v2f b, v8f c) {
    return __builtin_amdgcn_wmma_f32_16x16x4_f32(false, a, false, b,
                                                 (short)0, c, false, false);
}

// -------------------------------------------------------------------------
// Kernel 1: chol_params = x @ W^T + b, scattered into square L (lower tri),
// ReLU on diagonal. Block = 128 batch rows x 64 P cols; 8 waves as 4x2,
// each wave a 32x32 subtile (4 f32 accumulators). Double-buffered LDS,
// float4/b128 staging, one barrier per 16-wide K step.
// -------------------------------------------------------------------------
__global__ __launch_bounds__(256)
void linear_tri_kernel(const float* __restrict__ x, const float* __restrict__ W,
                       const float* __restrict__ bias, float* __restrict__ L) {
    __shared__ float As[2][128][LDS_PAD];
    __shared__ float Bs[2][64][LDS_PAD];

    const int tid  = threadIdx.x;
    const int lane = tid & 31;
    const int wave = tid >> 5;
    const int blockM = blockIdx.y * 128;         // batch rows
    const int blockN = blockIdx.x * 64;          // P columns
    const int waveM  = (wave & 3) * 32;
    const int waveN  = (wave >> 2) * 32;

    const int ldRow = tid >> 2;                  // 0..63
    const int ldCol = (tid & 3) * 4;             // 0,4,8,12

    const float* gA0 = x + (long long)(blockM + ldRow) * IN_K + ldCol;
    const float* gA1 = x + (long long)(blockM + 64 + ldRow) * IN_K + ldCol;
    const float* gB0 = W + (long long)(blockN + ldRow) * IN_K + ldCol;

    v8f acc00 = {}, acc01 = {}, acc10 = {}, acc11 = {};

    const int m_a = lane & 15;                   // fragment row (M or N)
    const int kq  = (lane >> 4) * 2;             // K pair select (0 or 2)

    // prologue: tile 0
    float4 ra0 = *(const float4*)(gA0);
    float4 ra1 = *(const float4*)(gA1);
    float4 rb0 = *(const float4*)(gB0);
    *(float4*)&As[0][ldRow][ldCol]      = ra0;
    *(float4*)&As[0][64 + ldRow][ldCol] = ra1;
    *(float4*)&Bs[0][ldRow][ldCol]      = rb0;
    __syncthreads();

    int cur = 0;
    for (int kb = 16; kb < IN_K; kb += 16) {
        ra0 = *(const float4*)(gA0 + kb);        // prefetch next tile
        ra1 = *(const float4*)(gA1 + kb);
        rb0 = *(const float4*)(gB0 + kb);
#pragma unroll
        for (int kk = 0; kk < 16; kk += 4) {
            v2f a0 = *(const v2f*)&As[cur][waveM + m_a][kk + kq];
            v2f a1 = *(const v2f*)&As[cur][waveM + 16 + m_a][kk + kq];
            v2f b0 = *(const v2f*)&Bs[cur][waveN + m_a][kk + kq];
            v2f b1 = *(const v2f*)&Bs[cur][waveN + 16 + m_a][kk + kq];
            acc00 = wmma4(a0, b0, acc00);
            acc01 = wmma4(a0, b1, acc01);
            acc10 = wmma4(a1, b0, acc10);
            acc11 = wmma4(a1, b1, acc11);
        }
        const int nxt = cur ^ 1;
        *(float4*)&As[nxt][ldRow][ldCol]      = ra0;
        *(float4*)&As[nxt][64 + ldRow][ldCol] = ra1;
        *(float4*)&Bs[nxt][ldRow][ldCol]      = rb0;
        __syncthreads();
        cur = nxt;
    }
#pragma unroll
    for (int kk = 0; kk < 16; kk += 4) {         // last tile
        v2f a0 = *(const v2f*)&As[cur][waveM + m_a][kk + kq];
        v2f a1 = *(const v2f*)&As[cur][waveM + 16 + m_a][kk + kq];
        v2f b0 = *(const v2f*)&Bs[cur][waveN + m_a][kk + kq];
        v2f b1 = *(const v2f*)&Bs[cur][waveN + 16 + m_a][kk + kq];
        acc00 = wmma4(a0, b0, acc00);
        acc01 = wmma4(a0, b1, acc01);
        acc10 = wmma4(a1, b0, acc10);
        acc11 = wmma4(a1, b1, acc11);
    }

    // Scatter-store: D layout = 8 VGPRs; lane<16: (M=r, N=lane), lane>=16: (M=r+8, N=lane-16)
    const int nloc = lane & 15;
    const int hi   = lane >> 4;
#pragma unroll
    for (int nb = 0; nb < 2; ++nb) {
        const int p = blockN + waveN + nb * 16 + nloc;
        int ti, tj;
        tri_inverse(p, ti, tj);
        const float bv   = bias[p];
        const bool  diag = (ti == tj);
        float* dst = L + (long long)ti * M_DIM + tj;
#pragma unroll
        for (int ma = 0; ma < 2; ++ma) {
            const v8f acc = nb ? (ma ? acc11 : acc01) : (ma ? acc10 : acc00);
#pragma unroll
            for (int r = 0; r < 8; ++r) {
                const int m = blockM + waveM + ma * 16 + hi * 8 + r;
                float v = acc[r] + bv;
                if (diag) v = fmaxf(v, 0.0f);
                dst[(long long)m * (M_DIM * M_DIM)] = v;
            }
        }
    }
}

// -------------------------------------------------------------------------
// Kernel 2: out[b] = L[b] @ L[b]^T (batched SYRK, full 256x256 output).
// Same 128x64 / 32x32-per-wave WMMA tiling. Upper triangle of L is never
// written; staging masks k>row components to 0, and the K loop is clamped
// to min(tileI+128, tileJ+64)  (k <= min(i,j): ~half the FLOPs).
// -------------------------------------------------------------------------
__global__ __launch_bounds__(256)
void syrk_kernel(const float* __restrict__ L, float* __restrict__ out) {
    __shared__ float As[2][128][LDS_PAD];
    __shared__ float Bs[2][64][LDS_PAD];

    const int tid   = threadIdx.x;
    const int lane  = tid & 31;
    const int wave  = tid >> 5;
    const int batch = blockIdx.y;
    const int tileI = (blockIdx.x & 1) * 128;
    const int tileJ = (blockIdx.x >> 1) * 64;

    const float* __restrict__ Lb = L + (long long)batch * (M_DIM * M_DIM);

    const int waveM = (wave & 3) * 32;
    const int waveN = (wave >> 2) * 32;
    const int ldRow = tid >> 2;                  // 0..63
    const int ldCol = (tid & 3) * 4;             // 0,4,8,12
    const int rA0 = tileI + ldRow;
    const int rA1 = tileI + 64 + ldRow;
    const int rB0 = tileJ + ldRow;

    auto ldmask = [&](int row, int k0) -> float4 {
        float4 v = *(const float4*)&Lb[row * M_DIM + k0];
        v.x = (k0     <= row) ? v.x : 0.0f;      // zero above the diagonal
        v.y = (k0 + 1 <= row) ? v.y : 0.0f;
        v.z = (k0 + 2 <= row) ? v.z : 0.0f;
        v.w = (k0 + 3 <= row) ? v.w : 0.0f;
        return v;
    };

    v8f acc00 = {}, acc01 = {}, acc10 = {}, acc11 = {};
    const int m_a = lane & 15;
    const int kq  = (lane >> 4) * 2;

    const int kmax = min(tileI + 128, tileJ + 64);

    float4 ra0 = ldmask(rA0, ldCol);
    float4 ra1 = ldmask(rA1, ldCol);
    float4 rb0 = ldmask(rB0, ldCol);
    *(float4*)&As[0][ldRow][ldCol]      = ra0;
    *(float4*)&As[0][64 + ldRow][ldCol] = ra1;
    *(float4*)&Bs[0][ldRow][ldCol]      = rb0;
    __syncthreads();

    int cur = 0;
    for (int kb = 16; kb < kmax; kb += 16) {
        ra0 = ldmask(rA0, kb + ldCol);
        ra1 = ldmask(rA1, kb + ldCol);
        rb0 = ldmask(rB0, kb + ldCol);
#pragma unroll
        for (int kk = 0; kk < 16; kk += 4) {
            v2f a0 = *(const v2f*)&As[cur][waveM + m_a][kk + kq];
            v2f a1 = *(const v2f*)&As[cur][waveM + 16 + m_a][kk + kq];
            v2f b0 = *(const v2f*)&Bs[cur][waveN + m_a][kk + kq];
            v2f b1 = *(const v2f*)&Bs[cur][waveN + 16 + m_a][kk + kq];
            acc00 = wmma4(a0, b0, acc00);
            acc01 = wmma4(a0, b1, acc01);
            acc10 = wmma4(a1, b0, acc10);
            acc11 = wmma4(a1, b1, acc11);
        }
        const int nxt = cur ^ 1;
        *(float4*)&As[nxt][ldRow][ldCol]      = ra0;
        *(float4*)&As[nxt][64 + ldRow][ldCol] = ra1;
        *(float4*)&Bs[nxt][ldRow][ldCol]      = rb0;
        __syncthreads();
        cur = nxt;
    }
#pragma unroll
    for (int kk = 0; kk < 16; kk += 4) {
        v2f a0 = *(const v2f*)&As[cur][waveM + m_a][kk + kq];
        v2f a1 = *(const v2f*)&As[cur][waveM + 16 + m_a][kk + kq];
        v2f b0 = *(const v2f*)&Bs[cur][waveN + m_a][kk + kq];
        v2f b1 = *(const v2f*)&Bs[cur][waveN + 16 + m_a][kk + kq];
        acc00 = wmma4(a0, b0, acc00);
        acc01 = wmma4(a0, b1, acc01);
        acc10 = wmma4(a1, b0, acc10);
        acc11 = wmma4(a1, b1, acc11);
    }

    const int nloc = lane & 15;
    const int hi   = lane >> 4;
    float* __restrict__ ob = out + (long long)batch * (M_DIM * M_DIM);
#pragma unroll
    for (int nb = 0; nb < 2; ++nb) {
        const int j = tileJ + waveN + nb * 16 + nloc;
#pragma unroll
        for (int ma = 0; ma < 2; ++ma) {
            const v8f acc = nb ? (ma ? acc11 : acc01) : (ma ? acc10 : acc00);
#pragma unroll
            for (int r = 0; r < 8; ++r) {
                const int i = tileI + waveM + ma * 16 + hi * 8 + r;
                ob[i * M_DIM + j] = acc[r];
            }
        }
    }
}

extern "C" void kernel_launch(void* const* d_in, const int* in_sizes, int n_in,
                              void* d_out, int out_size, void* d_ws, size_t ws_size,
                              hipStream_t stream) {
    (void)in_sizes; (void)n_in; (void)out_size; (void)ws_size;
    const float* x    = (const float*)d_in[0];   // [2048, 1024]
    const float* W    = (const float*)d_in[1];   // [32896, 1024]
    const float* bias = (const float*)d_in[2];   // [32896]
    float* out  = (float*)d_out;                 // [2048, 256, 256]
    float* Lbuf = (float*)d_ws;                  // [2048, 256, 256] lower-tri (512 MB)

    // GEMM + triangular scatter: grid (P/64, B/128) = (514, 16)
    dim3 gridB(P_DIM / 64, BATCH / 128);
    linear_tri_kernel<<<gridB, 256, 0, stream>>>(x, W, bias, Lbuf);

    // Batched SYRK: grid (2x4 tiles of 128x64, 2048 batches)
    dim3 gridC(8, BATCH);
    syrk_kernel<<<gridC, 256, 0, stream>>>(Lbuf, out);
}